// DISK_695784702679
// MI455X (gfx1250) — compile-verified
//
#include <hip/hip_runtime.h>
#include <stdint.h>
#include <math.h>

// ---------------------------------------------------------------------------
// DISK keypoint extraction for MI455X (gfx1250, wave32).
// Selection-logic bound problem (~5MB mandatory traffic @ 23.3TB/s => ~us).
//  - NMS tile staging via Tensor Data Mover (tensor_load_to_lds) for interior
//    blocks, per-lane global_load_async_to_lds_b32 for border blocks.
//  - Deterministic exact top-k via 8-level radix select on unique 64-bit keys.
//  - Descriptor L2 norms via V_WMMA_F32_16X16X4_F32 (Gram diagonal).
// ---------------------------------------------------------------------------

typedef float v2f __attribute__((ext_vector_type(2)));
typedef float v8f __attribute__((ext_vector_type(8)));
typedef unsigned int u32x4 __attribute__((ext_vector_type(4)));
typedef unsigned int u32x8 __attribute__((ext_vector_type(8)));

#define TILE   32
#define HALO   4
#define LW     (TILE + 2 * HALO)   // 40
#define SORT_N 2048

struct SelState {
  unsigned long long prefix;
  unsigned int rem;
  unsigned int pad0;
  unsigned long long threshold;
};

// Monotone float -> orderable u32 (larger float => larger uint).
__device__ __forceinline__ unsigned ordenc(float f) {
  unsigned u = __float_as_uint(f);
  return (u & 0x80000000u) ? ~u : (u | 0x80000000u);
}
__device__ __forceinline__ float orddec(unsigned o) {
  return (o & 0x80000000u) ? __uint_as_float(o & 0x7FFFFFFFu)
                           : __uint_as_float(~o);
}

__device__ __forceinline__ float v8f_get(v8f v, int i) {
  switch (i) {
    case 0: return v[0]; case 1: return v[1]; case 2: return v[2]; case 3: return v[3];
    case 4: return v[4]; case 5: return v[5]; case 6: return v[6]; default: return v[7];
  }
}

// --------------------------- init ------------------------------------------
__global__ void disk_init_kernel(unsigned int* counters, unsigned int* hist) {
  if (threadIdx.x < 4)   counters[threadIdx.x] = 0u;
  if (threadIdx.x < 256) hist[threadIdx.x]     = 0u;
}

// ------------------- NMS detect + candidate append + hist0 -----------------
__global__ void disk_detect_kernel(const float* __restrict__ feat,
                                   const int* __restrict__ p_ow,
                                   const int* __restrict__ p_oh,
                                   const int* __restrict__ p_win,
                                   int H, int W, int D,
                                   unsigned long long* __restrict__ cand,
                                   unsigned int* __restrict__ counters,
                                   unsigned int* __restrict__ hist,
                                   unsigned int cap) {
  __shared__ float tile[LW * LW];
  __shared__ unsigned int lh[256];
  const float* heat = feat + (size_t)D * H * W;
  const int ow = *p_ow, oh = *p_oh, win = *p_win;
  int r = win >> 1;
  if (r > HALO) r = HALO;
  if (r < 0)    r = 0;
  const int gx0 = blockIdx.x * TILE, gy0 = blockIdx.y * TILE;

  if (threadIdx.x < 256) lh[threadIdx.x] = 0u;

  const bool interior = (gx0 >= HALO) && (gy0 >= HALO) &&
                        (gx0 + TILE + HALO <= W) && (gy0 + TILE + HALO <= H);
  if (interior) {
    // ---- Tensor Data Mover: DMA the whole 40x40 tile into LDS, one op. ----
    if (threadIdx.x < 32) {                 // wave 0 only (uniform branch)
      unsigned lds_base = (unsigned)(uintptr_t)(void*)tile;   // addr[31:0] = LDS offset
      unsigned long long gaddr =
          (unsigned long long)(uintptr_t)(heat +
              ((size_t)(gy0 - HALO) * W + (gx0 - HALO)));
      u32x4 g0;
      g0[0] = 1u;                                          // count=1 (valid D#)
      g0[1] = lds_base;                                    // lds_addr
      g0[2] = (unsigned)gaddr;                             // global_addr[31:0]
      g0[3] = ((unsigned)(gaddr >> 32) & 0x01FFFFFFu)      // global_addr[56:32]
              | (2u << 30);                                // type=2 ("image")
      u32x8 g1;
      g1[0] = (2u << 16);                                  // data_size=4B
      g1[1] = ((unsigned)W & 0xFFFFu) << 16;               // tensor_dim0[15:0]
      g1[2] = (((unsigned)W >> 16) & 0xFFFFu)              // tensor_dim0[31:16]
              | (((unsigned)H & 0xFFFFu) << 16);           // tensor_dim1[15:0]
      g1[3] = (((unsigned)H >> 16) & 0xFFFFu)              // tensor_dim1[31:16]
              | ((unsigned)LW << 16);                      // tile_dim0 = 40
      g1[4] = (unsigned)LW;                                // tile_dim1 = 40 (tile_dim2=0)
      g1[5] = (unsigned)W;                                 // tensor_dim0_stride[31:0]
      g1[6] = 0u;                                          // stride hi / dim1_stride lo
      g1[7] = 0u;
      asm volatile("tensor_load_to_lds %0, %1"
                   :: "s"(g0), "s"(g1) : "memory");
      __builtin_amdgcn_s_wait_tensorcnt(0);
    }
    __syncthreads();
  } else {
    // ---- Border: prefill -inf (SAME padding), then async global->LDS. ----
    for (int i = threadIdx.x; i < LW * LW; i += blockDim.x) tile[i] = -INFINITY;
    __syncthreads();
    for (int i = threadIdx.x; i < LW * LW; i += blockDim.x) {
      int ly = i / LW, lx = i - ly * LW;
      int gy = gy0 + ly - HALO, gx = gx0 + lx - HALO;
      if (gx >= 0 && gx < W && gy >= 0 && gy < H) {
        unsigned lds_off = (unsigned)(uintptr_t)(void*)&tile[i];
        unsigned goff = ((unsigned)gy * (unsigned)W + (unsigned)gx) * 4u;
        asm volatile("global_load_async_to_lds_b32 %0, %1, %2"
                     :: "v"(lds_off), "v"(goff), "s"(heat) : "memory");
      }
    }
    asm volatile("s_wait_asynccnt 0" ::: "memory");
    __syncthreads();
  }

  const int tx = threadIdx.x & 15, ty = threadIdx.x >> 4;
  for (int s = 0; s < 4; ++s) {
    int lx = tx * 2 + (s & 1), ly = ty * 2 + (s >> 1);
    int gx = gx0 + lx, gy = gy0 + ly;
    float v = tile[(ly + HALO) * LW + (lx + HALO)];
    bool keep = (v > 0.0f) && (gx <= ow - 1) && (gy <= oh - 1) && (gx < W) && (gy < H);
    if (keep) {
      float m = -INFINITY;
      for (int dy = -r; dy <= r; ++dy) {
        const float* row = &tile[(ly + HALO + dy) * LW + (lx + HALO)];
        for (int dx = -r; dx <= r; ++dx) m = fmaxf(m, row[dx]);
      }
      if (v == m) {                                  // local max of its window
        unsigned idx = (unsigned)(gy * W + gx);
        unsigned o = ordenc(v);
        // unique composite: high score first, then lower idx first
        unsigned long long comp =
            ((unsigned long long)o << 32) | (unsigned)(~idx);
        unsigned pos = atomicAdd(&counters[0], 1u);
        if (pos < cap) cand[pos] = comp;
        atomicAdd(&lh[o >> 24], 1u);
      }
    }
  }
  __syncthreads();
  if (threadIdx.x < 256) {
    unsigned c = lh[threadIdx.x];
    if (c) atomicAdd(&hist[threadIdx.x], c);
  }
}

// ------------------- radix-select refine histogram -------------------------
__global__ void disk_hist_refine_kernel(const unsigned long long* __restrict__ cand,
                                        const unsigned int* __restrict__ counters,
                                        unsigned int* __restrict__ hist,
                                        const SelState* __restrict__ st,
                                        int shift, unsigned int cap) {
  unsigned n = counters[0];
  if (n > cap) n = cap;
  unsigned long long pre = st->prefix;
  unsigned stride = gridDim.x * blockDim.x;
  for (unsigned i = blockIdx.x * blockDim.x + threadIdx.x; i < n; i += stride) {
    unsigned long long c = cand[i];
    if ((c >> (shift + 8)) == pre)
      atomicAdd(&hist[(unsigned)(c >> shift) & 0xFFu], 1u);
  }
}

// ------------------- radix-select scan (1 block) ---------------------------
__global__ void disk_scan_kernel(unsigned int* hist, SelState* st,
                                 unsigned int K, int pass) {
  __shared__ unsigned int sh[256];
  if (threadIdx.x < 256) sh[threadIdx.x] = hist[threadIdx.x];
  __syncthreads();
  if (threadIdx.x == 0) {
    unsigned rem = (pass == 0) ? K : st->rem;
    unsigned long long prefix = (pass == 0) ? 0ull : st->prefix;
    unsigned cum = 0;
    int b = 255;
    for (; b > 0; --b) {
      unsigned h = sh[b];
      if (cum + h >= rem) break;
      cum += h;
    }
    unsigned long long np = (prefix << 8) | (unsigned long long)(unsigned)b;
    st->prefix = np;
    st->rem = rem - cum;
    if (pass == 7) st->threshold = np;   // exact K-th largest composite key
  }
  __syncthreads();
  if (threadIdx.x < 256) hist[threadIdx.x] = 0u;  // ready for next pass
}

// ------------------- compact winners (exactly K keys >= T) -----------------
__global__ void disk_compact_kernel(const unsigned long long* __restrict__ cand,
                                    unsigned int* __restrict__ counters,
                                    const SelState* __restrict__ st,
                                    unsigned long long* __restrict__ sel,
                                    unsigned int cap, unsigned int selcap) {
  unsigned n = counters[0];
  if (n > cap) n = cap;
  unsigned long long T = st->threshold;
  unsigned stride = gridDim.x * blockDim.x;
  for (unsigned i = blockIdx.x * blockDim.x + threadIdx.x; i < n; i += stride) {
    unsigned long long c = cand[i];
    if (c >= T && c != 0ull) {
      unsigned pos = atomicAdd(&counters[1], 1u);
      if (pos < selcap) sel[pos] = c;
    }
  }
}

// ------------------- bitonic sort (desc) + decode outputs ------------------
__global__ void disk_sort_output_kernel(const unsigned long long* __restrict__ sel,
                                        const unsigned int* __restrict__ counters,
                                        float* __restrict__ out,
                                        unsigned int* __restrict__ idx_sorted,
                                        int K, int W, int D) {
  __shared__ unsigned long long a[SORT_N];
  unsigned n = counters[1];
  if (n > (unsigned)K) n = (unsigned)K;
  for (int i = threadIdx.x; i < SORT_N; i += blockDim.x)
    a[i] = (i < (int)n) ? sel[i] : 0ull;
  __syncthreads();

  for (int size = 2; size <= SORT_N; size <<= 1) {
    for (int stride = size >> 1; stride > 0; stride >>= 1) {
      int t = threadIdx.x;                       // 1024 threads == SORT_N/2
      int i = 2 * t - (t & (stride - 1));
      int j = i + stride;
      bool down = ((i & size) == 0);             // descending overall
      unsigned long long x = a[i], y = a[j];
      if (down ? (x < y) : (x > y)) { a[i] = y; a[j] = x; }
      __syncthreads();
    }
  }

  for (int i = threadIdx.x; i < K; i += blockDim.x) {
    unsigned long long c = a[i];
    float score; unsigned idx; float valid;
    if (c == 0ull) {
      score = -INFINITY; idx = 0xFFFFFFFFu; valid = 0.0f;
    } else {
      unsigned o = (unsigned)(c >> 32);
      score = orddec(o);
      idx = ~(unsigned)c;
      valid = (isfinite(score)) ? 1.0f : 0.0f;
    }
    unsigned ix = (idx == 0xFFFFFFFFu) ? 0u : idx;
    out[2 * i + 0] = (float)(ix % (unsigned)W);          // x
    out[2 * i + 1] = (float)(ix / (unsigned)W);          // y
    out[2 * (size_t)K + i] = score;                      // scores
    out[3 * (size_t)K + (size_t)K * D + i] = valid;      // valid
    idx_sorted[i] = idx;
  }
}

// ------------- descriptor gather + WMMA Gram-diagonal L2 normalize ---------
// One wave handles 16 keypoints. Per-lane A layout for V_WMMA_F32_16X16X4_F32:
//   lanes 0-15 : M=lane,    VGPR0=K0, VGPR1=K1
//   lanes 16-31: M=lane-16, VGPR0=K2, VGPR1=K3
// Feeding the same registers for A and B yields D = A*A^T (Gram); the
// diagonal gives exact f32 squared norms, accumulated over 32 K-chunks.
__global__ void disk_desc_kernel(const float* __restrict__ feat,
                                 const unsigned int* __restrict__ idx_sorted,
                                 float* __restrict__ out,
                                 int K, int D, int HW) {
  __shared__ float snorm[8 * 16];
  const int lane = threadIdx.x & 31;
  const int wave = threadIdx.x >> 5;
  const int tile = blockIdx.x * (blockDim.x >> 5) + wave;
  const int kp_local = lane & 15;
  const int half = lane >> 4;               // 0: K0/K1 lanes, 1: K2/K3 lanes
  const int kp = tile * 16 + kp_local;

  unsigned idx = (kp < K) ? idx_sorted[kp] : 0xFFFFFFFFu;
  bool ok = idx < (unsigned)HW;
  unsigned safe = ok ? idx : 0u;

  v8f c = {0.f, 0.f, 0.f, 0.f, 0.f, 0.f, 0.f, 0.f};
  const int nchunk = D >> 2;                // 32 for D=128
  for (int k = 0; k < nchunk; ++k) {
    int ch0 = 4 * k + 2 * half;
    __builtin_prefetch(&feat[(size_t)(ch0 + 4) * HW + safe], 0, 1);
    float e0 = feat[(size_t)ch0 * HW + safe];
    float e1 = feat[(size_t)(ch0 + 1) * HW + safe];
    if (!ok) { e0 = 0.f; e1 = 0.f; }
    v2f a; a.x = e0; a.y = e1;
    // D(16x16,f32) += A(16x4,f32) * A^T(4x16,f32)
    c = __builtin_amdgcn_wmma_f32_16x16x4_f32(
        /*neg_a=*/false, a, /*neg_b=*/false, a,
        /*c_mod=*/(short)0, c, /*reuse_a=*/false, /*reuse_b=*/false);
  }

  // Diagonal (M==N) lives at: lane m (m<8) in VGPR m; lane m+16 (m>=8) in VGPR m-8.
  if (lane < 8)        snorm[wave * 16 + lane]        = v8f_get(c, lane);
  else if (lane >= 24) snorm[wave * 16 + (lane - 16)] = v8f_get(c, lane - 24);
  __syncthreads();

  float ss  = snorm[wave * 16 + kp_local];
  float nrm = sqrtf(fmaxf(ss, 0.0f));
  float inv = 1.0f / fmaxf(nrm, 1e-12f);

  if (kp < K) {
    size_t obase = 3 * (size_t)K + (size_t)kp * D;   // d section, row-major
    int ch_lo = (D >> 1) * half, ch_hi = ch_lo + (D >> 1);
    for (int ch = ch_lo; ch < ch_hi; ++ch) {
      float v = ok ? feat[(size_t)ch * HW + idx] : 0.0f;
      out[obase + ch] = v * inv;
    }
  }
}

// ---------------------------------------------------------------------------
extern "C" void kernel_launch(void* const* d_in, const int* in_sizes, int n_in,
                              void* d_out, int out_size, void* d_ws, size_t ws_size,
                              hipStream_t stream) {
  const float* feat = (const float*)d_in[0];
  const int* p_ow  = (const int*)d_in[1];
  const int* p_oh  = (const int*)d_in[2];
  const int* p_win = (const int*)d_in[4];

  const int H = 1024, W = 1024;
  const int HW = H * W;
  int D = in_sizes[0] / HW - 1;            // 128
  if (D < 1) D = 1;
  int K = out_size / (D + 4);              // 2 + 1 + D + 1 floats per keypoint
  if (K > SORT_N) K = SORT_N;

  // workspace layout (byte offsets; d_ws is 256B aligned)
  char* ws = (char*)d_ws;
  unsigned int* counters = (unsigned int*)(ws + 0);          //   64 B
  unsigned int* hist     = (unsigned int*)(ws + 64);         // 1024 B
  SelState*     st       = (SelState*)(ws + 1088);           //   64 B
  unsigned long long* sel = (unsigned long long*)(ws + 1152);        // 32 KB
  unsigned int* idx_sorted = (unsigned int*)(ws + 1152 + 32768);     //  8 KB
  unsigned long long* cand = (unsigned long long*)(ws + 1152 + 32768 + 8192);
  size_t fixed = 1152 + 32768 + 8192;
  unsigned int cap = 0;
  if (ws_size > fixed + 8) cap = (unsigned)((ws_size - fixed) / 8);
  if (cap > (unsigned)HW) cap = (unsigned)HW;

  disk_init_kernel<<<1, 256, 0, stream>>>(counters, hist);

  dim3 dgrid(W / TILE, H / TILE);
  disk_detect_kernel<<<dgrid, 256, 0, stream>>>(feat, p_ow, p_oh, p_win,
                                                H, W, D, cand, counters, hist, cap);

  disk_scan_kernel<<<1, 256, 0, stream>>>(hist, st, (unsigned)K, 0);
  for (int p = 1; p <= 7; ++p) {
    int shift = 64 - 8 * (p + 1);          // 48, 40, ..., 0
    disk_hist_refine_kernel<<<256, 256, 0, stream>>>(cand, counters, hist, st,
                                                     shift, cap);
    disk_scan_kernel<<<1, 256, 0, stream>>>(hist, st, (unsigned)K, p);
  }

  disk_compact_kernel<<<256, 256, 0, stream>>>(cand, counters, st, sel, cap, 4096u);

  disk_sort_output_kernel<<<1, SORT_N / 2, 0, stream>>>(sel, counters,
                                                        (float*)d_out, idx_sorted,
                                                        K, W, D);

  int tiles  = (K + 15) / 16;
  int blocks = (tiles + 7) / 8;            // 8 waves (tiles) per 256-thread block
  disk_desc_kernel<<<blocks, 256, 0, stream>>>(feat, idx_sorted, (float*)d_out,
                                               K, D, HW);
}